// GATNet_9208409883146
// MI455X (gfx1250) — compile-verified
//
#include <hip/hip_runtime.h>
#include <math.h>

typedef __attribute__((ext_vector_type(2))) float v2f;
typedef __attribute__((ext_vector_type(8))) float v8f;

#define N_NODES 20000
#define N_EDGES 160000
#define FDIM    128
#define HEADS   4
#define C1      128
#define C2      256
#define D1      512   // HEADS*C1
#define NGRAPH  64
#define NCLASS  10

#define BK   32      // k-chunk
#define LDA  36      // BK + 4  (A LDS stride; conflict-free for WMMA A reads)
#define LDB  272     // 256 + 16 (B LDS stride; half-waves hit disjoint banks)

// ---------------------------------------------------------------------------
// float atomic max via int/uint ordering trick (valid for finite + -inf init)
// ---------------------------------------------------------------------------
__device__ __forceinline__ void atomicMaxF(float* addr, float v) {
  if (v >= 0.0f) atomicMax((int*)addr, __float_as_int(v));
  else           atomicMin((unsigned int*)addr, (unsigned int)__float_as_int(v));
}

__global__ void fill_f32(float* __restrict__ p, float v, int n) {
  int i = blockIdx.x * blockDim.x + threadIdx.x;
  if (i < n) p[i] = v;
}

// ---------------------------------------------------------------------------
// fp32 GEMM via V_WMMA_F32_16X16X4_F32, LDS double-buffered.
//   C[M,N] = A[M,K] * B[K,N], row-major. grid = (N/256, M/32), block = 256.
//   8 waves: 2(M) x 4(N); each wave computes 16x64 via 4 v8f accumulators.
// WMMA fragment layout (ISA 7.12.2):
//   A: lane l -> A[l&15, k0 + (l>>4)*2 + {0,1}]
//   B: lane l -> B[k0 + (l>>4)*2 + {0,1}, l&15]
//   C/D: vgpr r of lane l -> C[r + 8*(l>>4), l&15]
// ---------------------------------------------------------------------------
template<int K>
__global__ __launch_bounds__(256, 2)
void gemm_wmma_f32(const float* __restrict__ A, const float* __restrict__ B,
                   float* __restrict__ C, int N) {
  constexpr int NCHUNK = K / BK;
  __shared__ float As[2][32 * LDA];
  __shared__ float Bs[2][BK * LDB];

  const int tid   = threadIdx.x;
  const int wave  = tid >> 5, lane = tid & 31;
  const int waveM = wave >> 2, waveN = wave & 3;
  const int blockM = blockIdx.y * 32;
  const int blockN = blockIdx.x * 256;

  // staging coordinates: A chunk 32x32 (1 float4/thread), B chunk 32x256 (8)
  const int arow = tid >> 3, acol = (tid & 7) << 2;

  float4 areg;
  float4 breg[8];

  auto gload = [&](int c) {
    areg = *(const float4*)&A[(size_t)(blockM + arow) * K + c * BK + acol];
#pragma unroll
    for (int i = 0; i < 8; ++i) {
      int q = tid + 256 * i;            // float4 index within 32x256 chunk
      int row = q >> 6, col4 = (q & 63) << 2;
      breg[i] = *(const float4*)&B[(size_t)(c * BK + row) * N + blockN + col4];
    }
  };
  auto lstore = [&](int buf) {
    *(float4*)&As[buf][arow * LDA + acol] = areg;
#pragma unroll
    for (int i = 0; i < 8; ++i) {
      int q = tid + 256 * i;
      int row = q >> 6, col4 = (q & 63) << 2;
      *(float4*)&Bs[buf][row * LDB + col4] = breg[i];
    }
  };

  gload(0);
  lstore(0);
  if (NCHUNK > 1) gload(1);
  __syncthreads();

  v8f acc[4] = {};
  const int mrow  = lane & 15;
  const int khalf = (lane >> 4) * 2;

  for (int c = 0; c < NCHUNK; ++c) {
    const int buf = c & 1;
    const float* as = &As[buf][waveM * 16 * LDA];
    const float* bs = &Bs[buf][waveN * 64];
#pragma unroll
    for (int k0 = 0; k0 < BK; k0 += 4) {
      const int koff = k0 + khalf;
      v2f a;
      a.x = as[mrow * LDA + koff];
      a.y = as[mrow * LDA + koff + 1];
#pragma unroll
      for (int j = 0; j < 4; ++j) {
        v2f b;
        b.x = bs[koff * LDB + j * 16 + mrow];
        b.y = bs[(koff + 1) * LDB + j * 16 + mrow];
        acc[j] = __builtin_amdgcn_wmma_f32_16x16x4_f32(
            false, a, false, b, (short)0, acc[j], false, false);
      }
    }
    if (c + 1 < NCHUNK) {
      lstore((c + 1) & 1);              // safe: last readers of this buffer
      if (c + 2 < NCHUNK) gload(c + 2); //   finished before the prior barrier
      __syncthreads();
    }
  }

  const int mbase = blockM + waveM * 16 + (lane >> 4) * 8;
#pragma unroll
  for (int j = 0; j < 4; ++j) {
    const int n = blockN + waveN * 64 + j * 16 + mrow;
#pragma unroll
    for (int r = 0; r < 8; ++r)
      C[(size_t)(mbase + r) * N + n] = acc[j][r];
  }
}

// ---------------------------------------------------------------------------
// Per-node attention dot products: asrc[n,h] = <h[n,h,:], att_src[h,:]>
// ---------------------------------------------------------------------------
__global__ __launch_bounds__(256)
void attn_dots(const float* __restrict__ h, const float* __restrict__ att_src,
               const float* __restrict__ att_dst, float* __restrict__ asrc,
               float* __restrict__ adst, int heads, int chead) {
  int node = blockIdx.x * (blockDim.x >> 5) + (threadIdx.x >> 5);
  int lane = threadIdx.x & 31;
  if (node >= N_NODES) return;
  const float* row = h + (size_t)node * heads * chead;
  for (int hh = 0; hh < heads; ++hh) {
    float s = 0.f, d = 0.f;
    for (int c = lane; c < chead; c += 32) {
      float v = row[hh * chead + c];
      s += v * att_src[hh * chead + c];
      d += v * att_dst[hh * chead + c];
    }
    for (int off = 16; off > 0; off >>= 1) {
      s += __shfl_down(s, off, 32);
      d += __shfl_down(d, off, 32);
    }
    if (lane == 0) {
      asrc[node * heads + hh] = s;
      adst[node * heads + hh] = d;
    }
  }
}

// directed edge i in [0, 2E+N): forward, reverse, then self loops
__device__ __forceinline__ void edge_uv(const int* __restrict__ ei, int i,
                                        int& u, int& v) {
  if (i < N_EDGES)          { u = ei[i];            v = ei[N_EDGES + i]; }
  else if (i < 2 * N_EDGES) { u = ei[i];            v = ei[i - N_EDGES]; }
  else                      { u = i - 2 * N_EDGES;  v = u; }
}

__device__ __forceinline__ float leaky02(float e) {
  return e > 0.f ? e : 0.2f * e;
}

template<int H>
__global__ __launch_bounds__(256)
void edge_max(const int* __restrict__ ei, const float* __restrict__ asrc,
              const float* __restrict__ adst, float* __restrict__ segmax) {
  int i = blockIdx.x * blockDim.x + threadIdx.x;
  if (i >= 2 * N_EDGES + N_NODES) return;
  int u, v; edge_uv(ei, i, u, v);
#pragma unroll
  for (int hh = 0; hh < H; ++hh) {
    float e = leaky02(asrc[u * H + hh] + adst[v * H + hh]);
    atomicMaxF(&segmax[v * H + hh], e);
  }
}

template<int H>
__global__ __launch_bounds__(256)
void edge_expsum(const int* __restrict__ ei, const float* __restrict__ asrc,
                 const float* __restrict__ adst, const float* __restrict__ segmax,
                 float* __restrict__ segsum) {
  int i = blockIdx.x * blockDim.x + threadIdx.x;
  if (i >= 2 * N_EDGES + N_NODES) return;
  int u, v; edge_uv(ei, i, u, v);
#pragma unroll
  for (int hh = 0; hh < H; ++hh) {
    float e = leaky02(asrc[u * H + hh] + adst[v * H + hh]);
    atomicAdd(&segsum[v * H + hh], expf(e - segmax[v * H + hh]));
  }
}

template<int H, int C>
__global__ __launch_bounds__(256)
void edge_scatter(const int* __restrict__ ei, const float* __restrict__ asrc,
                  const float* __restrict__ adst, const float* __restrict__ segmax,
                  const float* __restrict__ segsum, const float* __restrict__ h,
                  float* __restrict__ out) {
  int wid  = blockIdx.x * (blockDim.x >> 5) + (threadIdx.x >> 5);
  int lane = threadIdx.x & 31;
  if (wid >= 2 * N_EDGES + N_NODES) return;
  int u, v; edge_uv(ei, wid, u, v);
#pragma unroll
  for (int hh = 0; hh < H; ++hh) {
    float e = leaky02(asrc[u * H + hh] + adst[v * H + hh]);
    float alpha = expf(e - segmax[v * H + hh]) / segsum[v * H + hh];
    const float* hr = h   + (size_t)u * (H * C) + hh * C;
    float*     orow = out + (size_t)v * (H * C) + hh * C;
    for (int c = lane; c < C; c += 32)
      atomicAdd(&orow[c], hr[c] * alpha);
  }
}

__global__ __launch_bounds__(256)
void bias_relu(float* __restrict__ x, const float* __restrict__ b, int n, int d) {
  int i = blockIdx.x * blockDim.x + threadIdx.x;
  if (i >= n * d) return;
  float v = x[i] + b[i % d];
  x[i] = v > 0.f ? v : 0.f;
}

// global max pool (adds output bias b2 on the fly); one wave per node
__global__ __launch_bounds__(256)
void pool_max(const float* __restrict__ out2, const float* __restrict__ b2,
              const int* __restrict__ batch, float* __restrict__ gpool) {
  int node = blockIdx.x * (blockDim.x >> 5) + (threadIdx.x >> 5);
  int lane = threadIdx.x & 31;
  if (node >= N_NODES) return;
  int g = batch[node];
  for (int c = lane; c < C2; c += 32)
    atomicMaxF(&gpool[g * C2 + c], out2[(size_t)node * C2 + c] + b2[c]);
}

// fc head + log_softmax; one block per graph
__global__ __launch_bounds__(64)
void mlp_head(const float* __restrict__ gpool, const float* __restrict__ fc1w,
              const float* __restrict__ fc1b, const float* __restrict__ fc2w,
              const float* __restrict__ fc2b, float* __restrict__ out) {
  __shared__ float gs[C2];
  __shared__ float t[64];
  __shared__ float logits[NCLASS];
  int g = blockIdx.x, tid = threadIdx.x;
  for (int c = tid; c < C2; c += 64) gs[c] = gpool[g * C2 + c];
  __syncthreads();
  float acc = fc1b[tid];
  for (int c = 0; c < C2; ++c) acc += gs[c] * fc1w[c * 64 + tid];
  t[tid] = acc > 0.f ? acc : 0.f;
  __syncthreads();
  if (tid < NCLASS) {
    float l = fc2b[tid];
    for (int j = 0; j < 64; ++j) l += t[j] * fc2w[j * NCLASS + tid];
    logits[tid] = l;
  }
  __syncthreads();
  if (tid == 0) {
    float m = logits[0];
    for (int k = 1; k < NCLASS; ++k) m = fmaxf(m, logits[k]);
    float s = 0.f;
    for (int k = 0; k < NCLASS; ++k) s += expf(logits[k] - m);
    float lse = m + logf(s);
    for (int k = 0; k < NCLASS; ++k) out[g * NCLASS + k] = logits[k] - lse;
  }
}

// ---------------------------------------------------------------------------
extern "C" void kernel_launch(void* const* d_in, const int* in_sizes, int n_in,
                              void* d_out, int out_size, void* d_ws, size_t ws_size,
                              hipStream_t stream) {
  const float* x     = (const float*)d_in[0];
  const int*   ei    = (const int*)  d_in[1];
  const int*   batch = (const int*)  d_in[2];
  const float* W1    = (const float*)d_in[3];
  const float* as1   = (const float*)d_in[4];
  const float* ad1   = (const float*)d_in[5];
  const float* b1    = (const float*)d_in[6];
  const float* W2    = (const float*)d_in[7];
  const float* as2   = (const float*)d_in[8];
  const float* ad2   = (const float*)d_in[9];
  const float* b2    = (const float*)d_in[10];
  const float* fc1w  = (const float*)d_in[11];
  const float* fc1b  = (const float*)d_in[12];
  const float* fc2w  = (const float*)d_in[13];
  const float* fc2b  = (const float*)d_in[14];
  float* out = (float*)d_out;

  // workspace carve-up (~125 MB; entire working set fits in 192 MB L2)
  float* p     = (float*)d_ws;
  float* h1    = p; p += (size_t)N_NODES * D1;
  float* asrc1 = p; p += N_NODES * HEADS;
  float* adst1 = p; p += N_NODES * HEADS;
  float* smax1 = p; p += N_NODES * HEADS;
  float* ssum1 = p; p += N_NODES * HEADS;
  float* out1  = p; p += (size_t)N_NODES * D1;
  float* h2    = p; p += (size_t)N_NODES * C2;
  float* asrc2 = p; p += N_NODES;
  float* adst2 = p; p += N_NODES;
  float* smax2 = p; p += N_NODES;
  float* ssum2 = p; p += N_NODES;
  float* out2  = p; p += (size_t)N_NODES * C2;
  float* gpool = p; p += NGRAPH * C2;

  const int TOT = 2 * N_EDGES + N_NODES;             // 340000 directed edges
  const int eblk = (TOT + 255) / 256;                // thread-per-edge grids
  const int ewav = (TOT + 7) / 8;                    // wave-per-edge grids
  const int nwav = (N_NODES + 7) / 8;                // wave-per-node grids

  // ---- GATConv 1 ----
  gemm_wmma_f32<FDIM><<<dim3(D1 / 256, N_NODES / 32), 256, 0, stream>>>(x, W1, h1, D1);
  attn_dots<<<nwav, 256, 0, stream>>>(h1, as1, ad1, asrc1, adst1, HEADS, C1);
  fill_f32<<<(N_NODES * HEADS + 255) / 256, 256, 0, stream>>>(smax1, -INFINITY, N_NODES * HEADS);
  fill_f32<<<(N_NODES * HEADS + 255) / 256, 256, 0, stream>>>(ssum1, 0.f, N_NODES * HEADS);
  fill_f32<<<(N_NODES * D1 + 255) / 256, 256, 0, stream>>>(out1, 0.f, N_NODES * D1);
  edge_max<HEADS><<<eblk, 256, 0, stream>>>(ei, asrc1, adst1, smax1);
  edge_expsum<HEADS><<<eblk, 256, 0, stream>>>(ei, asrc1, adst1, smax1, ssum1);
  edge_scatter<HEADS, C1><<<ewav, 256, 0, stream>>>(ei, asrc1, adst1, smax1, ssum1, h1, out1);
  bias_relu<<<(N_NODES * D1 + 255) / 256, 256, 0, stream>>>(out1, b1, N_NODES, D1);

  // ---- GATConv 2 ----
  gemm_wmma_f32<D1><<<dim3(C2 / 256, N_NODES / 32), 256, 0, stream>>>(out1, W2, h2, C2);
  attn_dots<<<nwav, 256, 0, stream>>>(h2, as2, ad2, asrc2, adst2, 1, C2);
  fill_f32<<<(N_NODES + 255) / 256, 256, 0, stream>>>(smax2, -INFINITY, N_NODES);
  fill_f32<<<(N_NODES + 255) / 256, 256, 0, stream>>>(ssum2, 0.f, N_NODES);
  fill_f32<<<(N_NODES * C2 + 255) / 256, 256, 0, stream>>>(out2, 0.f, N_NODES * C2);
  edge_max<1><<<eblk, 256, 0, stream>>>(ei, asrc2, adst2, smax2);
  edge_expsum<1><<<eblk, 256, 0, stream>>>(ei, asrc2, adst2, smax2, ssum2);
  edge_scatter<1, C2><<<ewav, 256, 0, stream>>>(ei, asrc2, adst2, smax2, ssum2, h2, out2);

  // ---- pool + head ----
  fill_f32<<<(NGRAPH * C2 + 255) / 256, 256, 0, stream>>>(gpool, -INFINITY, NGRAPH * C2);
  pool_max<<<nwav, 256, 0, stream>>>(out2, b2, batch, gpool);
  mlp_head<<<NGRAPH, 64, 0, stream>>>(gpool, fc1w, fc1b, fc2w, fc2b, out);
}